// ReGroupConv2D_25847113188000
// MI455X (gfx1250) — compile-verified
//
#include <hip/hip_runtime.h>

// Problem constants (match reference)
#define B_N   64
#define CIN   256
#define COUT  256
#define G_N   1024

// Tiling
#define KC      32                    // K chunk
#define STRF    36                    // padded LDS row stride in floats (32 + 4) -> 144B rows, 16B aligned
#define XFL     (B_N * STRF)          // 2304 floats = 9216 B
#define WFL     (COUT * STRF)         // 9216 floats = 36864 B
#define STAGEF  (XFL + WFL)           // 11520 floats = 46080 B per stage
#define LDS_BYTES (2 * STAGEF * 4)    // 92160 B double-buffered
#define NCHUNK  (CIN / KC)            // 8

typedef __attribute__((ext_vector_type(2))) float v2f;
typedef __attribute__((ext_vector_type(8))) float v8f;

extern __shared__ float ldsbuf[];

__launch_bounds__(256)
__global__ void regroupconv_wmma_f32(const float* __restrict__ x,
                                     const float* __restrict__ W,
                                     const float* __restrict__ bias,
                                     float* __restrict__ out) {
    const int g    = blockIdx.x;          // one group per block
    const int tid  = threadIdx.x;         // 256 threads = 8 waves
    const int lane  = tid & 31;
    const int wave  = tid >> 5;
    const int laneN = lane & 15;          // N / M-row selector within tile
    const int laneH = lane >> 4;          // 0/1 half-wave
    const int mtile = wave >> 1;          // 0..3   (b-tile of 16)
    const int nhalf = wave & 1;           // 0..1   (which 8 o-tiles)

    // ---- async-load work decomposition (chunk-invariant per-thread pieces) ----
    const unsigned brow0 = (unsigned)(tid >> 5);   // 0..7  (x row group)
    const unsigned xkk   = (unsigned)(tid & 31);   // 0..31 (k within chunk)
    const unsigned wo0   = (unsigned)(tid >> 3);   // 0..31 (W row group)
    const unsigned wkg   = (unsigned)(tid & 7);    // 0..7  (k 4-float group)

    const float* wg = W + (size_t)g * (COUT * CIN);  // saddr base for this group's W

    // LDS byte base of the dynamic segment (low 32 bits of generic shared addr == LDS offset)
    const unsigned ldsBase = (unsigned)(size_t)(&ldsbuf[0]);

    // Issue one K-chunk of async global->LDS copies (16 instructions per wave).
    auto issue_chunk = [&](int kc, int sbuf) {
        const unsigned xl = ldsBase + (unsigned)sbuf * (STAGEF * 4u);
        const unsigned wl = xl + XFL * 4u;
        const unsigned k0 = (unsigned)(kc * KC);
        // X chunk: 64 rows x 32 k, 1 float per lane (strided source, L2-absorbed)
#pragma unroll
        for (int j = 0; j < 8; ++j) {
            unsigned brow = brow0 + 8u * (unsigned)j;
            unsigned goff = (brow * CIN + (k0 + xkk)) * (G_N * 4u) + (unsigned)g * 4u;
            unsigned loff = xl + (brow * STRF + xkk) * 4u;
            asm volatile("global_load_async_to_lds_b32 %0, %1, %2"
                         :: "v"(loff), "v"(goff), "s"(x) : "memory");
        }
        // W chunk: 256 rows x 32 k, 16B per lane, fully coalesced
#pragma unroll
        for (int j = 0; j < 8; ++j) {
            unsigned o    = wo0 + 32u * (unsigned)j;
            unsigned goff = (o * CIN + (k0 + wkg * 4u)) * 4u;
            unsigned loff = wl + (o * STRF + wkg * 4u) * 4u;
            asm volatile("global_load_async_to_lds_b128 %0, %1, %2"
                         :: "v"(loff), "v"(goff), "s"(wg) : "memory");
        }
    };

    // ---- accumulators: 8 o-tiles x (16x16 f32 = 8 VGPRs); init with bias ----
    v8f acc[8];
#pragma unroll
    for (int t = 0; t < 8; ++t) {
        float bv = bias[(size_t)g * COUT + (size_t)((nhalf * 8 + t) * 16 + laneN)];
        acc[t] = (v8f){bv, bv, bv, bv, bv, bv, bv, bv};
    }

    issue_chunk(0, 0);

    const int koff = laneH * 2;  // K pair owned by this half-wave (ISA 16x4 / 4x16 f32 layout)

    for (int kc = 0; kc < NCHUNK; ++kc) {
        if (kc + 1 < NCHUNK) {
            issue_chunk(kc + 1, (kc + 1) & 1);
            asm volatile("s_wait_asynccnt 16" ::: "memory");  // chunk kc landed (in-order per wave)
        } else {
            asm volatile("s_wait_asynccnt 0" ::: "memory");
        }
        __syncthreads();  // all waves' portions of chunk kc visible

        const float* xs = ldsbuf + (kc & 1) * STAGEF;
        const float* ws = xs + XFL;

#pragma unroll
        for (int ks = 0; ks < KC / 4; ++ks) {
            const int kl = ks * 4 + koff;
            // A fragment: X[b = mtile*16 + laneN, k = kl, kl+1]
            v2f a = *(const v2f*)(xs + (mtile * 16 + laneN) * STRF + kl);
#pragma unroll
            for (int t = 0; t < 8; ++t) {
                // B fragment: W[o = ntile*16 + laneN, k = kl, kl+1]  (B is KxN with N=o)
                v2f b = *(const v2f*)(ws + ((nhalf * 8 + t) * 16 + laneN) * STRF + kl);
                acc[t] = __builtin_amdgcn_wmma_f32_16x16x4_f32(
                    /*neg_a=*/false, a, /*neg_b=*/false, b,
                    /*c_mod=*/(short)0, acc[t],
                    /*reuse_a=*/false, /*reuse_b=*/false);
            }
        }
        __syncthreads();  // reads of this buffer done before it is refilled
    }

    // ---- store: out[b, o, g]; C/D layout: VGPR r -> M = r + 8*laneH, N = laneN ----
#pragma unroll
    for (int t = 0; t < 8; ++t) {
        const int o = (nhalf * 8 + t) * 16 + laneN;
#pragma unroll
        for (int r = 0; r < 8; ++r) {
            const int b = mtile * 16 + laneH * 8 + r;
            out[((size_t)b * COUT + (size_t)o) * G_N + (size_t)g] = acc[t][r];
        }
    }
}

extern "C" void kernel_launch(void* const* d_in, const int* in_sizes, int n_in,
                              void* d_out, int out_size, void* d_ws, size_t ws_size,
                              hipStream_t stream) {
    (void)in_sizes; (void)n_in; (void)d_ws; (void)ws_size; (void)out_size;
    const float* x    = (const float*)d_in[0];
    const float* W    = (const float*)d_in[1];
    const float* bias = (const float*)d_in[2];
    float* out        = (float*)d_out;

    // Allow >64KB dynamic LDS (no-op if already permitted); host-side, capture-safe.
    static_cast<void>(hipFuncSetAttribute((const void*)regroupconv_wmma_f32,
                                          hipFuncAttributeMaxDynamicSharedMemorySize,
                                          LDS_BYTES));

    regroupconv_wmma_f32<<<G_N, 256, LDS_BYTES, stream>>>(x, W, bias, out);
}